// LetterFilter_57234734187125
// MI455X (gfx1250) — compile-verified
//
#include <hip/hip_runtime.h>
#include <stdint.h>

typedef float v2f __attribute__((ext_vector_type(2)));
typedef float v8f __attribute__((ext_vector_type(8)));

#define Bn    2
#define Ln    64
#define Kn    16
#define Hn    256
#define PADn  8
#define HPn   272          // padded image 272x272
#define HHn   257          // conv output 257x257
#define EPSf  0.5f
#define ITERS 20
#define SPAT  (HHn * HHn)                 // 66049
#define LM_ELEMS ((size_t)Bn * Ln * SPAT) // 8454272
#define OUT2_ELEMS (Bn * HPn * HPn)       // 147968

// ---------------- workspace layout (bytes) ----------------
constexpr size_t LM_BYTES   = LM_ELEMS * 4;                    // 33,817,088
constexpr size_t FILLED_OFF = LM_BYTES;                        // Bn*SPAT ints
constexpr size_t SLOTS_OFF  = FILLED_OFF + (size_t)Bn * SPAT * 4; // %8==0
constexpr size_t REC_OFF    = SLOTS_OFF + (size_t)ITERS * Bn * 8; // 5 ints/rec
constexpr size_t HLIST_OFF  = REC_OFF + (size_t)ITERS * Bn * 5 * 4;
constexpr size_t HCNT_OFF   = HLIST_OFF + (size_t)Bn * ITERS * 4 * 4;
constexpr size_t DONE_OFF   = HCNT_OFF + Bn * 4;

// ---------------- init: zero lh region + bookkeeping ----------------
__global__ void lf_init_k(float* out_lh, int* filled, unsigned long long* slots,
                          int* hit_rec, int* hit_count, int* done) {
    size_t t = (size_t)blockIdx.x * blockDim.x + threadIdx.x;
    size_t stride = (size_t)gridDim.x * blockDim.x;
    for (size_t i = t; i < LM_ELEMS; i += stride) out_lh[i] = 0.f;
    for (size_t i = t; i < (size_t)Bn * SPAT; i += stride) filled[i] = 0;
    for (size_t i = t; i < (size_t)ITERS * Bn; i += stride) slots[i] = 0ull;
    for (size_t i = t; i < (size_t)ITERS * Bn * 5; i += stride) hit_rec[i] = 0;
    if (t == 0) { hit_count[0] = 0; hit_count[1] = 0; *done = 0; }
}

// ---------------- WMMA fp32 implicit-GEMM conv ----------------
// out[b,ch,py,px] = conv_bias[ch] + sum_k W[ch,k] * xpad[py+ky, px+kx]
// GEMM per wave: M=16 channels, N=16 spatial cols (one row py), K=256 in 64x K=4.
// A (16x4 f32): lane t -> m=t%16; vgpr0=K(kk), vgpr1=K(kk+1), kk = t<16?0:2.
// B (4x16 f32): lane t -> n=t%16; vgpr0=K(kk) row, vgpr1=K(kk+1) row.
// D (16x16 f32): lane t, vgpr r -> M=r+(t<16?0:8), N=t%16.
__global__ __launch_bounds__(128)
void lf_conv_wmma_k(const float* __restrict__ img, const float* __restrict__ Wg,
                    const float* __restrict__ cbias, float* __restrict__ lm) {
    const int b   = blockIdx.z;
    const int px0 = blockIdx.x * 16;   // 17 tiles
    const int py0 = blockIdx.y * 8;    // 33 row blocks of 8
    const int tid = threadIdx.x;

    // Stage replication-padded window: rows py0..py0+22, cols px0..px0+30
    __shared__ float sx[23][32];
    const float* imb = img + (size_t)b * Hn * Hn;
    for (int i = tid; i < 23 * 32; i += 128) {
        int r = i >> 5, c = i & 31;
        int gy = py0 + r - PADn; gy = gy < 0 ? 0 : (gy > Hn - 1 ? Hn - 1 : gy);
        int gx = px0 + c - PADn; gx = gx < 0 ? 0 : (gx > Hn - 1 ? Hn - 1 : gx);
        sx[r][c] = imb[gy * Hn + gx];
    }
    __syncthreads();

    const int lane = tid & 31;
    const int l0   = (tid >> 5) * 16;           // wave's channel group
    const int m16  = lane & 15;
    const int kk   = (lane < 16) ? 0 : 2;
    const float* wbase = Wg + (size_t)(l0 + m16) * 256;

    v8f accs[8];
#pragma unroll
    for (int i = 0; i < 8; ++i)
#pragma unroll
        for (int j = 0; j < 8; ++j) accs[i][j] = 0.f;

    for (int ky = 0; ky < 16; ++ky) {
        v2f a[4];
#pragma unroll
        for (int q = 0; q < 4; ++q)   // 8B-aligned: kk even
            a[q] = *(const v2f*)(wbase + ky * 16 + q * 4 + kk);
#pragma unroll
        for (int pl = 0; pl < 8; ++pl) {
            const float* xr = &sx[pl + ky][0];
#pragma unroll
            for (int q = 0; q < 4; ++q) {
                int bc = m16 + q * 4 + kk;
                v2f bv; bv.x = xr[bc]; bv.y = xr[bc + 1];
                accs[pl] = __builtin_amdgcn_wmma_f32_16x16x4_f32(
                    false, a[q], false, bv, (short)0, accs[pl], false, false);
            }
        }
    }

    const int px = px0 + m16;
    if (px < HHn) {
#pragma unroll
        for (int pl = 0; pl < 8; ++pl) {
            int py = py0 + pl;
            if (py >= HHn) break;
#pragma unroll
            for (int r = 0; r < 8; ++r) {
                int ch = l0 + r + ((lane < 16) ? 0 : 8);
                lm[(((size_t)b * Ln + ch) * HHn + py) * HHn + px] =
                    accs[pl][r] + cbias[ch];
            }
        }
    }
}

// ---------------- per-batch argmax via packed atomicMax ----------------
__global__ void lf_argmax_k(const float* __restrict__ lm, unsigned long long* slots,
                            const int* done, int iter) {
    if (*done) return;
    const int b = blockIdx.y;
    __shared__ unsigned long long sred[256];
    const float* p = lm + (size_t)b * Ln * SPAT;
    unsigned long long best = 0ull;
    const int n = Ln * SPAT;
    for (int i = blockIdx.x * blockDim.x + threadIdx.x; i < n;
         i += gridDim.x * blockDim.x) {
        unsigned u = __float_as_uint(p[i]);
        unsigned ord = (u & 0x80000000u) ? ~u : (u | 0x80000000u);
        // ~i in low bits => ties resolve to smallest index (argmax semantics)
        unsigned long long pk = ((unsigned long long)ord << 32) | (unsigned)(~i);
        if (pk > best) best = pk;
    }
    sred[threadIdx.x] = best;
    __syncthreads();
    for (int s = 128; s > 0; s >>= 1) {
        if (threadIdx.x < s && sred[threadIdx.x + s] > sred[threadIdx.x])
            sred[threadIdx.x] = sred[threadIdx.x + s];
        __syncthreads();
    }
    if (threadIdx.x == 0) atomicMax(&slots[iter * Bn + b], sred[0]);
}

// ---------------- pick: threshold, deposit hit, fill mask ----------------
__global__ void lf_pick_k(const float* __restrict__ lm,
                          const unsigned long long* __restrict__ slots,
                          int* filled, int* hit_rec, int* hlist, int* hit_count,
                          int* done, float* out_lh, int iter) {
    if (threadIdx.x != 0 || blockIdx.x != 0) return;
    int any = 0;
    const int was_done = *done;
    for (int b = 0; b < Bn; ++b) {
        int* rec = hit_rec + (iter * Bn + b) * 5;
        rec[0] = 0;
        if (was_done) continue;
        unsigned long long pk = slots[iter * Bn + b];
        unsigned idx = ~(unsigned)(pk & 0xffffffffu);
        if (idx >= (unsigned)(Ln * SPAT)) continue;   // empty slot guard
        float v = lm[(size_t)b * Ln * SPAT + idx];
        if (v > EPSf) {
            int l = idx / SPAT;
            int rem = idx - l * SPAT;
            int py = rem / HHn, px = rem - py * HHn;
            rec[0] = 1; rec[1] = l; rec[2] = py; rec[3] = px;
            rec[4] = __float_as_int(v);
            out_lh[(((size_t)b * Ln + l) * HHn + py) * HHn + px] = v;
            filled[b * SPAT + rem] = 1;
            int c = hit_count[b]++;
            int* hl = hlist + (b * ITERS + c) * 4;
            hl[0] = l; hl[1] = py; hl[2] = px; hl[3] = __float_as_int(v);
            any = 1;
        }
    }
    if (!was_done && !any) *done = 1;
}

// ---------------- rank-1 Gram patch update: lm -= v*G, re-apply filled ----
__global__ void lf_update_k(float* __restrict__ lm, const float* __restrict__ Wg,
                            const int* __restrict__ filled,
                            const int* __restrict__ hit_rec, int iter) {
    const int b = blockIdx.y;
    const int* rec = hit_rec + (iter * Bn + b) * 5;
    if (rec[0] == 0) return;
    const int lp = rec[1], py = rec[2], px = rec[3];
    const float v = __int_as_float(rec[4]);
    const int lq = blockIdx.x;

    __shared__ float swq[256], swp[256];
    for (int i = threadIdx.x; i < 256; i += blockDim.x) {
        swq[i] = Wg[lq * 256 + i];
        swp[i] = Wg[lp * 256 + i];
    }
    __syncthreads();

    for (int o = threadIdx.x; o < 31 * 31; o += blockDim.x) {
        int dy = o / 31 - 15, dx = o % 31 - 15;
        int i0 = dy < 0 ? -dy : 0, i1 = dy > 0 ? 16 - dy : 16;
        int j0 = dx < 0 ? -dx : 0, j1 = dx > 0 ? 16 - dx : 16;
        float g = 0.f;
        for (int i = i0; i < i1; ++i)
            for (int j = j0; j < j1; ++j)
                g += swq[i * 16 + j] * swp[(i + dy) * 16 + (j + dx)];
        int qy = py + dy, qx = px + dx;
        if (qy >= 0 && qy < HHn && qx >= 0 && qx < HHn) {
            int pos = qy * HHn + qx;
            size_t off = ((size_t)b * Ln + lq) * SPAT + pos;
            lm[off] = filled[b * SPAT + pos] ? 0.f : (lm[off] - v * g);
        }
    }
}

// ---------------- final conv_transpose(letter_hits) by hit scatter -------
__global__ void lf_scatter_k(float* __restrict__ out2, const float* __restrict__ Wg,
                             const float* __restrict__ tbias,
                             const int* __restrict__ hlist,
                             const int* __restrict__ hit_count) {
    int t = blockIdx.x * blockDim.x + threadIdx.x;
    if (t >= OUT2_ELEMS) return;
    int b = t / (HPn * HPn);
    int rem = t - b * HPn * HPn;
    int y = rem / HPn, x = rem - y * HPn;
    float acc = tbias[0];
    int n = hit_count[b];
    for (int h = 0; h < n; ++h) {
        const int* hl = hlist + (b * ITERS + h) * 4;
        int dy = y - hl[1], dx = x - hl[2];
        if (dy >= 0 && dy < Kn && dx >= 0 && dx < Kn)
            acc += __int_as_float(hl[3]) * Wg[hl[0] * 256 + dy * 16 + dx];
    }
    out2[t] = acc;
}

extern "C" void kernel_launch(void* const* d_in, const int* in_sizes, int n_in,
                              void* d_out, int out_size, void* d_ws, size_t ws_size,
                              hipStream_t stream) {
    (void)in_sizes; (void)n_in; (void)out_size; (void)ws_size;
    const float* img   = (const float*)d_in[0];  // [2,1,256,256]
    const float* Wg    = (const float*)d_in[1];  // [64,1,16,16]
    const float* cbias = (const float*)d_in[2];  // [64]
    const float* tbias = (const float*)d_in[3];  // [1]

    float* out_lh = (float*)d_out;               // [2,64,257,257]
    float* out2   = out_lh + LM_ELEMS;           // [2,1,272,272]

    char* ws = (char*)d_ws;
    float* lm                 = (float*)ws;
    int* filled               = (int*)(ws + FILLED_OFF);
    unsigned long long* slots = (unsigned long long*)(ws + SLOTS_OFF);
    int* hit_rec              = (int*)(ws + REC_OFF);
    int* hlist                = (int*)(ws + HLIST_OFF);
    int* hit_count            = (int*)(ws + HCNT_OFF);
    int* done                 = (int*)(ws + DONE_OFF);

    lf_init_k<<<1024, 256, 0, stream>>>(out_lh, filled, slots, hit_rec, hit_count, done);
    lf_conv_wmma_k<<<dim3(17, 33, Bn), 128, 0, stream>>>(img, Wg, cbias, lm);

    for (int it = 0; it < ITERS; ++it) {
        lf_argmax_k<<<dim3(256, Bn), 256, 0, stream>>>(lm, slots, done, it);
        lf_pick_k<<<1, 32, 0, stream>>>(lm, slots, filled, hit_rec, hlist,
                                        hit_count, done, out_lh, it);
        lf_update_k<<<dim3(Ln, Bn), 256, 0, stream>>>(lm, Wg, filled, hit_rec, it);
    }

    lf_scatter_k<<<(OUT2_ELEMS + 255) / 256, 256, 0, stream>>>(out2, Wg, tbias,
                                                               hlist, hit_count);
}